// EmbeddingEncoder_25297357373919
// MI455X (gfx1250) — compile-verified
//
#include <hip/hip_runtime.h>
#include <hip/hip_bf16.h>
#include <math.h>

// ---------------------------------------------------------------------------
// EmbeddingEncoder for MI455X (gfx1250): fp32 WMMA (v_wmma_f32_16x16x4_f32)
// + async global->LDS staging (global_load_async_to_lds_b128 / ASYNCcnt).
// B=128, S=384, D=96, H=4, K=7, L=4.  M = B*S = 49152 rows.
// Workspace: 7 * M*96 + M*48 floats  ~= 141.6 MB
// ---------------------------------------------------------------------------

typedef float v8f __attribute__((ext_vector_type(8)));
typedef float v2f __attribute__((ext_vector_type(2)));

#define DM 96
#define BB 128
#define SS 384
#define MM (BB * SS)

__device__ __forceinline__ v8f wmma_f32(v2f a, v2f b, v8f c) {
    return __builtin_amdgcn_wmma_f32_16x16x4_f32(false, a, false, b, (short)0, c,
                                                 false, false);
}

// ------------------------------ positional encoding ------------------------
__global__ void posenc_kernel(const float* __restrict__ in, float* __restrict__ x,
                              int total) {
    int idx = blockIdx.x * blockDim.x + threadIdx.x;
    if (idx >= total) return;
    int c = idx % DM;
    int s = (idx / DM) % SS;
    float e = (c & 1) ? 2.0f * (float)(c + 1) : 2.0f * (float)c;
    float ang = (float)s * __powf(10000.0f, -e * (1.0f / 96.0f));
    float pe = (c & 1) ? __cosf(ang) : __sinf(ang);
    x[idx] = in[idx] * 9.797958971f + pe;  // sqrt(96)
}

// ------------------------------ layernorm (wave per row) -------------------
__global__ void ln_kernel(const float* __restrict__ x, const float* __restrict__ g,
                          const float* __restrict__ b, float* __restrict__ out,
                          int rows) {
    int row = blockIdx.x * (blockDim.x >> 5) + (threadIdx.x >> 5);
    if (row >= rows) return;
    int lane = threadIdx.x & 31;
    const float* xr = x + (long)row * DM;
    float v0 = xr[lane], v1 = xr[lane + 32], v2 = xr[lane + 64];
    float s = v0 + v1 + v2;
    for (int o = 16; o > 0; o >>= 1) s += __shfl_xor(s, o, 32);
    float mean = s * (1.0f / 96.0f);
    float d0 = v0 - mean, d1 = v1 - mean, d2 = v2 - mean;
    float vs = d0 * d0 + d1 * d1 + d2 * d2;
    for (int o = 16; o > 0; o >>= 1) vs += __shfl_xor(vs, o, 32);
    float inv = rsqrtf(vs * (1.0f / 96.0f) + 1e-5f);
    float* orow = out + (long)row * DM;
    orow[lane]      = d0 * inv * g[lane]      + b[lane];
    orow[lane + 32] = d1 * inv * g[lane + 32] + b[lane + 32];
    orow[lane + 64] = d2 * inv * g[lane + 64] + b[lane + 64];
}

// ------------------------------ depthwise conv (K=7, SAME) -----------------
__global__ void dwconv_kernel(const float* __restrict__ in,
                              const float* __restrict__ w,
                              const float* __restrict__ bias,
                              float* __restrict__ out, int total) {
    int idx = blockIdx.x * blockDim.x + threadIdx.x;
    if (idx >= total) return;
    int c = idx % DM;
    int s = (idx / DM) % SS;
    int b = idx / (DM * SS);
    float acc = bias[c];
#pragma unroll
    for (int t = 0; t < 7; ++t) {
        int ss = s + t - 3;
        if (ss >= 0 && ss < SS)
            acc += in[((long)b * SS + ss) * DM + c] * w[c * 7 + t];
    }
    out[idx] = acc;
}

// ------------------------------ WMMA GEMM (templated) ----------------------
// out[m,n] = act( sum_k X[m,k] * Bmat[k,n] + bias[n] ) (+ residual[m,n])
// Bmat[k,n] = TRANSW ? W[n*K+k] : W[k*N+n].
// W staged in LDS in fragment-ready layout: one ds_load_b64 per B-fragment.
// Each wave owns one 16-row m-tile and all N/16 n-tiles (6 accumulators for
// N=96) so one A-fragment load feeds NT independent WMMAs.
template <int N, int K, int ACT, bool TRANSW, bool HASBIAS, bool HASRES>
__global__ __launch_bounds__(256) void gemm_kernel(
    const float* __restrict__ X, const float* __restrict__ W,
    const float* __restrict__ bias, const float* __restrict__ residual,
    float* __restrict__ out, int M) {
    // frag layout: wlds[(((k0/4)*2 + hi)*N + n)*2 + j] = Bmat[k0+2*hi+j][n]
    __shared__ __align__(16) float wlds[K * N];
    if (TRANSW) {
        // read float4 along k (contiguous rows of W), two v2f LDS stores
        for (int i4 = threadIdx.x; i4 < N * (K / 4); i4 += 256) {
            int n = i4 / (K / 4), kq = i4 % (K / 4);
            float4 w4 = *(const float4*)(W + (size_t)n * K + kq * 4);
            v2f lo = {w4.x, w4.y};
            v2f hi2 = {w4.z, w4.w};
            *(v2f*)(wlds + ((kq * 2 + 0) * N + n) * 2) = lo;
            *(v2f*)(wlds + ((kq * 2 + 1) * N + n) * 2) = hi2;
        }
    } else {
        // read float4 along n (coalesced), scatter 4 LDS words
        for (int i4 = threadIdx.x; i4 < K * (N / 4); i4 += 256) {
            int k = i4 / (N / 4), n = (i4 % (N / 4)) * 4;
            float4 w4 = *(const float4*)(W + (size_t)k * N + n);
            int h = (k & 3) >> 1, jb = k & 1;
            int basei = (((k >> 2) * 2 + h) * N) * 2 + jb;
            wlds[basei + (n + 0) * 2] = w4.x;
            wlds[basei + (n + 1) * 2] = w4.y;
            wlds[basei + (n + 2) * 2] = w4.z;
            wlds[basei + (n + 3) * 2] = w4.w;
        }
    }
    __syncthreads();

    const int lane = threadIdx.x & 31;
    const int wave = threadIdx.x >> 5;
    const int hi = lane >> 4;
    const int l16 = lane & 15;
    const int mtiles = M >> 4;
    const int waves_total = (gridDim.x * 256) >> 5;
    constexpr int NT = N / 16;

    for (int mt = blockIdx.x * 8 + wave; mt < mtiles; mt += waves_total) {
        const float* arow = X + (size_t)(mt * 16 + l16) * K;
        int nmt = mt + waves_total;
        if (nmt < mtiles)
            __builtin_prefetch(X + (size_t)(nmt * 16 + l16) * K, 0, 0);

        v8f acc[NT] = {};
#pragma unroll
        for (int k0 = 0; k0 < K; k0 += 4) {
            v2f a = *(const v2f*)(arow + k0 + 2 * hi);
#pragma unroll
            for (int nt = 0; nt < NT; ++nt) {
                v2f bf = *(const v2f*)(wlds +
                                       (((k0 >> 2) * 2 + hi) * N + nt * 16 + l16) * 2);
                acc[nt] = wmma_f32(a, bf, acc[nt]);
            }
        }
#pragma unroll
        for (int nt = 0; nt < NT; ++nt) {
            int n = nt * 16 + l16;
            float bi = HASBIAS ? bias[n] : 0.0f;
#pragma unroll
            for (int r = 0; r < 8; ++r) {
                size_t m = (size_t)mt * 16 + r + 8 * hi;
                float v = acc[nt][r] + bi;
                if (ACT == 1) v = fmaxf(v, 0.0f);
                if (ACT == 2) v = 1.0f / (1.0f + __expf(-v));
                if (HASRES) v += residual[m * N + n];
                out[m * N + n] = v;
            }
        }
    }
}

// ------------------------------ fused flash attention ----------------------
// One head at a time.  Q,K,V,ctx: [M x 96] with row = b*S + s.
// grid = (3, 128); 8 waves/block, each wave owns one 16-row q-tile.
// K tile staged via CDNA5 async copy (global_load_async_to_lds_b128,
// ASYNCcnt); V tile staged transposed with float4 loads so PV B-fragments
// are contiguous ds_load_b64.  Q fragments cached in registers.
__global__ __launch_bounds__(256) void attn_kernel(
    const float* __restrict__ Q, const float* __restrict__ Km,
    const float* __restrict__ V, const unsigned char* __restrict__ mask,
    float* __restrict__ ctx) {
    __shared__ __align__(16) float kt[16 * DM];   // kt[key][k]
    __shared__ __align__(16) float vt[DM * 16];   // vt[col][key]
    __shared__ __align__(16) float pls[8 * 256];  // per-wave 16x16 P staging
    const int lane = threadIdx.x & 31;
    const int wave = threadIdx.x >> 5;
    const int hi = lane >> 4;
    const int l16 = lane & 15;
    const int qt = blockIdx.x * 8 + wave;  // q-tile 0..23
    const int b = blockIdx.y;
    const int base = b * SS;
    float* p = pls + wave * 256;
    const unsigned kt_lds = (unsigned)(size_t)kt;

    // cache Q fragments in registers: 24 k-steps
    const float* qrow = Q + (size_t)(base + qt * 16 + l16) * DM;
    v2f qa[DM / 4];
#pragma unroll
    for (int k0 = 0; k0 < DM; k0 += 4) qa[k0 >> 2] = *(const v2f*)(qrow + k0 + 2 * hi);

    v8f O[6] = {};
    float mi[8], li[8];
    bool mrow[8];
#pragma unroll
    for (int r = 0; r < 8; ++r) {
        mi[r] = -1e30f;
        li[r] = 0.0f;
        mrow[r] = mask[base + qt * 16 + r + 8 * hi] != 0;
    }
    const float scale = 9.797958971f;  // sqrt(96)

    for (int j = 0; j < SS / 16; ++j) {
        __syncthreads();  // previous iteration's kt/vt reads complete
        // ---- async-copy K tile (6 KB = 384 x b128), tracked by ASYNCcnt ----
        {
            unsigned long long ga =
                (unsigned long long)(size_t)(Km + (size_t)(base + j * 16) * DM) +
                (unsigned long long)threadIdx.x * 16u;
            unsigned la = kt_lds + threadIdx.x * 16u;
            asm volatile("global_load_async_to_lds_b128 %0, %1, off"
                         :: "v"(la), "v"(ga) : "memory");
            if (threadIdx.x < 128) {
                ga += 256ull * 16u;
                la += 256u * 16u;
                asm volatile("global_load_async_to_lds_b128 %0, %1, off"
                             :: "v"(la), "v"(ga) : "memory");
            }
        }
        // ---- stage V tile transposed (float4 reads, 4x b32 LDS scatter) ----
        {
            int i = threadIdx.x;
            {
                int r = i / 24, c = (i % 24) * 4;
                float4 v4 = *(const float4*)(V + (size_t)(base + j * 16 + r) * DM + c);
                vt[(c + 0) * 16 + r] = v4.x;
                vt[(c + 1) * 16 + r] = v4.y;
                vt[(c + 2) * 16 + r] = v4.z;
                vt[(c + 3) * 16 + r] = v4.w;
            }
            if (i < 128) {
                int idx = i + 256;
                int r = idx / 24, c = (idx % 24) * 4;
                float4 v4 = *(const float4*)(V + (size_t)(base + j * 16 + r) * DM + c);
                vt[(c + 0) * 16 + r] = v4.x;
                vt[(c + 1) * 16 + r] = v4.y;
                vt[(c + 2) * 16 + r] = v4.z;
                vt[(c + 3) * 16 + r] = v4.w;
            }
        }
        asm volatile("s_wait_asynccnt 0x0" ::: "memory");
        __syncthreads();

        // ---- S_tile = Q (16x96) * K^T (96x16) ----
        v8f sc = {};
#pragma unroll
        for (int k0 = 0; k0 < DM; k0 += 4) {
            v2f bf = *(const v2f*)(kt + l16 * DM + k0 + 2 * hi);
            sc = wmma_f32(qa[k0 >> 2], bf, sc);
        }
        // ---- online softmax (rows r+8*hi live in 16-lane groups) ----
#pragma unroll
        for (int r = 0; r < 8; ++r) {
            float s = sc[r] * scale;
            if (mrow[r]) s = -1e30f;
            float mx = s;
            for (int o = 1; o < 16; o <<= 1) mx = fmaxf(mx, __shfl_xor(mx, o, 32));
            float newm = fmaxf(mi[r], mx);
            float alpha = __expf(mi[r] - newm);
            float pv = __expf(s - newm);
            sc[r] = pv;
            float rs = pv;
            for (int o = 1; o < 16; o <<= 1) rs += __shfl_xor(rs, o, 32);
            li[r] = li[r] * alpha + rs;
            mi[r] = newm;
#pragma unroll
            for (int t = 0; t < 6; ++t) O[t][r] = O[t][r] * alpha;
        }
        // ---- P: C-layout -> A-layout via per-wave LDS region ----
#pragma unroll
        for (int r = 0; r < 8; ++r) p[(r + 8 * hi) * 16 + l16] = sc[r];
        // (same-wave DS ops are in order; no cross-wave access to p)
        // ---- O += P (16x16) * V (16x96) ----
#pragma unroll
        for (int k0 = 0; k0 < 16; k0 += 4) {
            v2f a = *(const v2f*)(p + l16 * 16 + k0 + 2 * hi);
#pragma unroll
            for (int nt = 0; nt < 6; ++nt) {
                v2f bf = *(const v2f*)(vt + (nt * 16 + l16) * 16 + k0 + 2 * hi);
                O[nt] = wmma_f32(a, bf, O[nt]);
            }
        }
    }
    // ---- normalize and store ctx ----
#pragma unroll
    for (int r = 0; r < 8; ++r) {
        float invl = 1.0f / li[r];
        float* orow = ctx + (size_t)(base + qt * 16 + r + 8 * hi) * DM;
#pragma unroll
        for (int t = 0; t < 6; ++t) orow[t * 16 + l16] = O[t][r] * invl;
    }
}

// ---------------------------------------------------------------------------
extern "C" void kernel_launch(void* const* d_in, const int* in_sizes, int n_in,
                              void* d_out, int out_size, void* d_ws, size_t ws_size,
                              hipStream_t stream) {
    const float* input      = (const float*)d_in[0];
    const unsigned char* mk = (const unsigned char*)d_in[1];
    const float* conv_dw    = (const float*)d_in[2];   // [4,96,1,7]
    const float* conv_dw_b  = (const float*)d_in[3];   // [4,96]
    const float* conv_pw    = (const float*)d_in[4];   // [4,96,96,1]
    const float* conv_pw_b  = (const float*)d_in[5];   // [4,96]
    const float* WQ         = (const float*)d_in[6];   // [4,96,96]
    const float* WK         = (const float*)d_in[7];
    const float* WV         = (const float*)d_in[8];
    const float* WO         = (const float*)d_in[9];   // [384,96]
    const float* ffn_w1     = (const float*)d_in[10];  // [96,48]
    const float* ffn_b1     = (const float*)d_in[11];
    const float* ffn_w2     = (const float*)d_in[12];  // [48,96]
    const float* ffn_b2     = (const float*)d_in[13];
    const float* ln_g       = (const float*)d_in[14];  // [6,96]
    const float* ln_b       = (const float*)d_in[15];
    float* out = (float*)d_out;

    float* x   = (float*)d_ws;        // M*96
    float* ln  = x   + (size_t)MM * DM;
    float* tmp = ln  + (size_t)MM * DM;
    float* q   = tmp + (size_t)MM * DM;
    float* k   = q   + (size_t)MM * DM;
    float* v   = k   + (size_t)MM * DM;
    float* ctx = v   + (size_t)MM * DM;
    float* t1  = ctx + (size_t)MM * DM; // M*48

    const int total = MM * DM;
    const int EB = 256, EG = (total + EB - 1) / EB;
    const int GB = 384;  // 384 blocks * 8 waves = 3072 = M/16 m-tiles

    // x = input*sqrt(96) + PE
    posenc_kernel<<<EG, EB, 0, stream>>>(input, x, total);

    // conv blocks: x = x + relu(pw(dw(LN(x))))
    for (int l = 0; l < 4; ++l) {
        ln_kernel<<<MM / 8, 256, 0, stream>>>(x, ln_g + l * DM, ln_b + l * DM, ln, MM);
        dwconv_kernel<<<EG, EB, 0, stream>>>(ln, conv_dw + l * DM * 7,
                                             conv_dw_b + l * DM, tmp, total);
        gemm_kernel<96, 96, 1, true, true, true><<<GB, 256, 0, stream>>>(
            tmp, conv_pw + l * DM * DM, conv_pw_b + l * DM, x, x, MM);
    }

    // attention (loop heads; x += ctx_h @ WO_h accumulates per head)
    ln_kernel<<<MM / 8, 256, 0, stream>>>(x, ln_g + 4 * DM, ln_b + 4 * DM, ln, MM);
    for (int h = 0; h < 4; ++h) {
        gemm_kernel<96, 96, 0, false, false, false><<<GB, 256, 0, stream>>>(
            ln, WQ + h * DM * DM, nullptr, nullptr, q, MM);
        gemm_kernel<96, 96, 0, false, false, false><<<GB, 256, 0, stream>>>(
            ln, WK + h * DM * DM, nullptr, nullptr, k, MM);
        gemm_kernel<96, 96, 0, false, false, false><<<GB, 256, 0, stream>>>(
            ln, WV + h * DM * DM, nullptr, nullptr, v, MM);
        dim3 ag(3, BB);
        attn_kernel<<<ag, 256, 0, stream>>>(q, k, v, mk, ctx);
        gemm_kernel<96, 96, 0, false, false, true><<<GB, 256, 0, stream>>>(
            ctx, WO + h * DM * DM, nullptr, x, x, MM);
    }

    // FFN: out = x + (sigmoid(LN(x) @ w1 + b1) @ w2 + b2)
    ln_kernel<<<MM / 8, 256, 0, stream>>>(x, ln_g + 5 * DM, ln_b + 5 * DM, ln, MM);
    gemm_kernel<48, 96, 2, false, true, false><<<GB, 256, 0, stream>>>(
        ln, ffn_w1, ffn_b1, nullptr, t1, MM);
    gemm_kernel<96, 48, 0, false, true, true><<<GB, 256, 0, stream>>>(
        t1, ffn_w2, ffn_b2, x, out, MM);
}